// SpectralConv2d_ssd_20263655702998
// MI455X (gfx1250) — compile-verified
//
#include <hip/hip_runtime.h>
#include <math.h>

#define DEV __device__ __forceinline__

typedef __attribute__((ext_vector_type(2))) float v2f;
typedef __attribute__((ext_vector_type(8))) float v8f;

DEV v8f wmma4(v2f a, v2f b, v8f c) {
  // V_WMMA_F32_16X16X4_F32: D = A(16x4,f32) * B(4x16,f32) + C(16x16,f32)
  return __builtin_amdgcn_wmma_f32_16x16x4_f32(false, a, false, b, (short)0, c, false, false);
}

// Async Global->LDS copy (gfx1250, ASYNCcnt-tracked). VDST = LDS byte address
// (low 32 bits of the generic shared pointer = wave-relative LDS address per ISA 10.2).
DEV void async_ld_b128(unsigned lds_addr, const void* gaddr) {
  asm volatile("global_load_async_to_lds_b128 %0, %1, off"
               :: "v"(lds_addr), "v"(gaddr) : "memory");
}
DEV void wait_async0() { asm volatile("s_wait_asynccnt 0x0" ::: "memory"); }
DEV unsigned lds_off(const void* p) { return (unsigned)(unsigned long long)p; }

// Problem dimensions (fixed by the reference)
constexpr int Bn  = 4;      // batch
constexpr int Cin = 64;     // input channels
constexpr int Hh  = 128, Ww = 128;
constexpr int Dd  = 128;    // 2*Cin
constexpr int Nn  = 64;     // state dim
constexpr int Wf  = 65;     // rfft bins
constexpr int Lf  = Hh * Wf;        // 8320 tokens
constexpr int CH  = 64;             // SSD chunk length
constexpr int NCH = Lf / CH;        // 130 chunks

// ---------------- serial 128-pt radix-2 FFT on bank-interleaved LDS ----------------
DEV void fft128_strided(float* re, float* im, int stride, float sign) {
  for (int i = 1; i < 127; ++i) {
    int j = (int)(__brev((unsigned)i) >> 25);   // 7-bit reversal
    if (j > i) {
      float tr = re[i*stride]; re[i*stride] = re[j*stride]; re[j*stride] = tr;
      float ti = im[i*stride]; im[i*stride] = im[j*stride]; im[j*stride] = ti;
    }
  }
  for (int len = 2; len <= 128; len <<= 1) {
    int half = len >> 1;
    float ang = sign * 6.28318530717958647692f / (float)len;
    for (int base = 0; base < 128; base += len) {
      for (int k = 0; k < half; ++k) {
        float s, c;
        __sincosf(ang * (float)k, &s, &c);
        int i0 = (base + k) * stride, i1 = (base + k + half) * stride;
        float xr = re[i1], xi = im[i1];
        float tr = c * xr - s * xi;
        float ti = c * xi + s * xr;
        float ur = re[i0], ui = im[i0];
        re[i0] = ur + tr; im[i0] = ui + ti;
        re[i1] = ur - tr; im[i1] = ui - ti;
      }
    }
  }
}

// K1: real FFT along W for every (b,c,h) row -> bins 0..64
__global__ __launch_bounds__(32) void k_rfft_rows(const float* __restrict__ x,
                                                  float* __restrict__ wre,
                                                  float* __restrict__ wim) {
  __shared__ float sre[128 * 32], sim[128 * 32];
  int t = threadIdx.x;
  int r = blockIdx.x * 32 + t;                 // (b,c,h) flat
  const float* row = x + (size_t)r * Ww;
  float* re = sre + t; float* im = sim + t;
  for (int i = 0; i < 128; ++i) { re[i*32] = row[i]; im[i*32] = 0.f; }
  fft128_strided(re, im, 32, -1.f);
  float* owr = wre + (size_t)r * Wf;
  float* owi = wim + (size_t)r * Wf;
  for (int k = 0; k <= 64; ++k) { owr[k] = re[k*32]; owi[k] = im[k*32]; }
}

// K2: complex FFT along H per (b,c,w) column, ortho scale, pack re/im into u[b, d, l]
__global__ __launch_bounds__(32) void k_fft_cols_fwd(const float* __restrict__ wre,
                                                     const float* __restrict__ wim,
                                                     float* __restrict__ u) {
  __shared__ float sre[128 * 32], sim[128 * 32];
  int q = blockIdx.x * 32 + threadIdx.x;       // (b,c,w)
  int w = q % Wf; int bc = q / Wf;
  int c = bc % Cin; int b = bc / Cin;
  float* re = sre + threadIdx.x; float* im = sim + threadIdx.x;
  const size_t base = (size_t)bc * Hh * Wf + w;
  for (int h = 0; h < 128; ++h) { re[h*32] = wre[base + (size_t)h*Wf]; im[h*32] = wim[base + (size_t)h*Wf]; }
  fft128_strided(re, im, 32, -1.f);
  const float sc = 1.f / 128.f;                // norm='ortho': 1/sqrt(H*W)
  for (int h = 0; h < 128; ++h) {
    size_t l = (size_t)h * Wf + w;
    u[((size_t)b * Dd + c)        * Lf + l] = re[h*32] * sc;
    u[((size_t)b * Dd + Cin + c)  * Lf + l] = im[h*32] * sc;
  }
}

// K3: LayerNorm over D=128 channels (+ /sqrt_n) in place; dt = softplus(w_dt . u + bias)
__global__ __launch_bounds__(32) void k_ln_dt(float* __restrict__ u,
                                              const float* __restrict__ nw,
                                              const float* __restrict__ nb,
                                              const float* __restrict__ wdt,
                                              const float* __restrict__ dtbias,
                                              float* __restrict__ dtb) {
  int idx = blockIdx.x * 32 + threadIdx.x;     // token index over (b,l), lanes = consecutive l
  int b = idx / Lf; int l = idx % Lf;
  const size_t base = (size_t)b * Dd * Lf + l;
  float s = 0.f, ss = 0.f;
  for (int d = 0; d < Dd; ++d) {
    float v = u[base + (size_t)d * Lf];
    s += v; ss += v * v;
  }
  float mu = s * (1.f / 128.f);
  float var = ss * (1.f / 128.f) - mu * mu;
  float rs = rsqrtf(var + 1e-5f);
  float acc = 0.f;
  for (int d = 0; d < Dd; ++d) {
    size_t o = base + (size_t)d * Lf;
    float v = ((u[o] - mu) * rs * nw[d] + nb[d]) * (1.f / 128.f);  // / sqrt_n
    u[o] = v;
    acc += wdt[d] * v;
  }
  float z = acc + dtbias[0];
  float dt = (z > 20.f) ? z : log1pf(__expf(z));   // softplus
  dtb[(size_t)b * Lf + l] = dt;
}

// K4: fused projection GEMM: rows 0..127 -> X (W_x), 128..191 -> B (W_B), 192..255 -> C (W_C)
__global__ __launch_bounds__(128) void k_proj_gemm(const float* __restrict__ u,
                                                   const float* __restrict__ Wx,
                                                   const float* __restrict__ Wb,
                                                   const float* __restrict__ Wc,
                                                   float* __restrict__ Xb,
                                                   float* __restrict__ Bb,
                                                   float* __restrict__ Cb) {
  __shared__ float Us[128 * 16];               // u-hat tile: [K=128][16 tokens]
  int tt = blockIdx.x % (Lf / 16);
  int b  = blockIdx.x / (Lf / 16);
  int l0 = tt * 16;
  int tid = threadIdx.x;
  const size_t ubase = (size_t)b * Dd * Lf + l0;
  __builtin_prefetch(u + ubase + (size_t)tid * Lf + 16, 0, 0);  // next token tile
  for (int j = 0; j < 16; ++j) Us[tid * 16 + j] = u[ubase + (size_t)tid * Lf + j];
  __syncthreads();
  int wave = tid >> 5, lane = tid & 31, ml = lane & 15, hi = lane >> 4, nl = ml;
  for (int mt = wave; mt < 16; mt += 4) {
    int rowA = mt * 16 + ml;
    const float* wrow = (rowA < 128) ? (Wx + (size_t)rowA * 128)
                     : (rowA < 192) ? (Wb + (size_t)(rowA - 128) * 128)
                                    : (Wc + (size_t)(rowA - 192) * 128);
    v8f acc = {0.f,0.f,0.f,0.f,0.f,0.f,0.f,0.f};
    for (int k = 0; k < 128; k += 4) {
      int ks = k + 2 * hi;
      v2f a; a.x = wrow[ks];            a.y = wrow[ks + 1];
      v2f bb; bb.x = Us[ks * 16 + nl];  bb.y = Us[(ks + 1) * 16 + nl];
      acc = wmma4(a, bb, acc);
    }
    int l = l0 + nl;
    for (int j = 0; j < 8; ++j) {
      int rowD = mt * 16 + j + 8 * hi;
      float v = acc[j];
      if (rowD < 128)      Xb[((size_t)b * Lf + l) * 128 + rowD] = v;
      else if (rowD < 192) Bb[((size_t)b * Lf + l) * 64 + (rowD - 128)] = v;
      else                 Cb[((size_t)b * Lf + l) * 64 + (rowD - 192)] = v;
    }
  }
}

// K5a: PARALLEL chunk work (no cross-chunk dependency): one block per (b, chunk).
//   G = C @ B^T; M~ = lowtri(exp(-a(cum_t-cum_s))*dt_s) o G;
//   Y_intra = M~ @ X + D_skip*X  -> Yb
//   dH      = X^T @ diag(w) B    -> per-chunk state increment (consumed by K5b)
//   Pvec, Ptot                    -> per-token / total chunk decay
__global__ __launch_bounds__(256) void k_chunk_intra(const float* __restrict__ Xb,
                                                     const float* __restrict__ Bb,
                                                     const float* __restrict__ Cb,
                                                     const float* __restrict__ dtb,
                                                     const float* __restrict__ Dskip,
                                                     const float* __restrict__ alog,
                                                     float* __restrict__ Yb,
                                                     float* __restrict__ dH,
                                                     float* __restrict__ Pvec,
                                                     float* __restrict__ Ptot) {
  extern __shared__ float sm[];
  float* Xc  = sm;                    // [64][129] tokens x D, padded
  float* Bc  = Xc + 64 * 129;         // [64][65]
  float* Cc  = Bc + 64 * 65;          // [64][65]
  float* Gm  = Cc + 64 * 65;          // [64][65]  G then M~
  float* cum = Gm + 64 * 65;          // [64]
  float* dts = cum + 64;              // [64]
  float* Pv  = dts + 64;              // [64]
  float* wv  = Pv + 64;               // [64]

  int blk = blockIdx.x;
  int b = blk / NCH, ch = blk % NCH;
  int l0 = ch * CH;
  int tid = threadIdx.x;
  int wave = tid >> 5, lane = tid & 31, ml = lane & 15, hi = lane >> 4, nl = ml;
  float aexp = __expf(alog[0]);

  // async-stage chunk tiles into LDS (16B per lane, rows are contiguous, pad not crossed)
  for (int i = tid; i < CH * 32; i += 256) {          // X: 64 x 128 floats
    int t = i >> 5, dg = (i & 31) * 4;
    async_ld_b128(lds_off(&Xc[t * 129 + dg]), Xb + ((size_t)b * Lf + l0 + t) * 128 + dg);
  }
  for (int i = tid; i < CH * 16; i += 256) {          // B, C: 64 x 64 floats each
    int t = i >> 4, dg = (i & 15) * 4;
    async_ld_b128(lds_off(&Bc[t * 65 + dg]), Bb + ((size_t)b * Lf + l0 + t) * 64 + dg);
    async_ld_b128(lds_off(&Cc[t * 65 + dg]), Cb + ((size_t)b * Lf + l0 + t) * 64 + dg);
  }
  if (tid < 64) dts[tid] = dtb[(size_t)b * Lf + l0 + tid];
  wait_async0();
  __syncthreads();

  if (tid == 0) {                                     // log-space decay prefix (no underflow)
    float s = 0.f;
    for (int t = 0; t < 64; ++t) { s += dts[t]; cum[t] = s; }
    float cT = cum[63];
    for (int t = 0; t < 64; ++t) {
      Pv[t] = __expf(-aexp * cum[t]);
      wv[t] = __expf(-aexp * (cT - cum[t])) * dts[t];
    }
  }
  // G = Cc @ Bc^T : 16 tiles of 16x16 (concurrent with the prefix; needs only Bc/Cc)
  for (int q = wave * 2; q < wave * 2 + 2; ++q) {
    int mt = q >> 2, nt = q & 3;
    v8f acc = {0.f,0.f,0.f,0.f,0.f,0.f,0.f,0.f};
    for (int k = 0; k < 64; k += 4) {
      int ks = k + 2 * hi;
      v2f a;  a.x  = Cc[(mt * 16 + ml) * 65 + ks];  a.y  = Cc[(mt * 16 + ml) * 65 + ks + 1];
      v2f bb; bb.x = Bc[(nt * 16 + nl) * 65 + ks];  bb.y = Bc[(nt * 16 + nl) * 65 + ks + 1];
      acc = wmma4(a, bb, acc);
    }
    for (int j = 0; j < 8; ++j)
      Gm[(mt * 16 + j + 8 * hi) * 65 + nt * 16 + nl] = acc[j];
  }
  __syncthreads();

  // masked decay matrix in place; also publish Pvec/Ptot
  for (int i = tid; i < 4096; i += 256) {
    int t = i >> 6, s = i & 63;
    float g = Gm[t * 65 + s];
    Gm[t * 65 + s] = (s <= t) ? __expf(-aexp * (cum[t] - cum[s])) * dts[s] * g : 0.f;
  }
  if (tid < 64) Pvec[(size_t)blk * 64 + tid] = Pv[tid];
  if (tid == 0) Ptot[blk] = Pv[63];
  __syncthreads();

  // Y_intra = M~ @ X  (+ D_skip * X)  : 4 x 8 tiles of 16x16, K = 64
  for (int q = wave * 4; q < wave * 4 + 4; ++q) {
    int mt = q >> 3, nt = q & 7;
    v8f acc = {0.f,0.f,0.f,0.f,0.f,0.f,0.f,0.f};
    for (int k = 0; k < 64; k += 4) {
      int ks = k + 2 * hi;
      v2f a;  a.x  = Gm[(mt * 16 + ml) * 65 + ks];  a.y  = Gm[(mt * 16 + ml) * 65 + ks + 1];
      v2f bb; bb.x = Xc[ks * 129 + nt * 16 + nl];   bb.y = Xc[(ks + 1) * 129 + nt * 16 + nl];
      acc = wmma4(a, bb, acc);
    }
    int dcol = nt * 16 + nl;
    float dsk = Dskip[dcol];
    for (int j = 0; j < 8; ++j) {
      int t = mt * 16 + j + 8 * hi;
      Yb[((size_t)b * Dd + dcol) * Lf + l0 + t] = acc[j] + dsk * Xc[t * 129 + dcol];
    }
  }

  // dH = X^T @ diag(wv) Bc : 8 x 4 tiles of 16x16, K = 64 (reads only, no barrier needed)
  for (int q = wave * 4; q < wave * 4 + 4; ++q) {
    int mt = q >> 2, nt = q & 3;
    v8f acc = {0.f,0.f,0.f,0.f,0.f,0.f,0.f,0.f};
    for (int k = 0; k < 64; k += 4) {
      int ks = k + 2 * hi;
      v2f a;  a.x  = Xc[ks * 129 + mt * 16 + ml];   a.y  = Xc[(ks + 1) * 129 + mt * 16 + ml];
      v2f bb; bb.x = wv[ks]     * Bc[ks * 65 + nt * 16 + nl];
              bb.y = wv[ks + 1] * Bc[(ks + 1) * 65 + nt * 16 + nl];
      acc = wmma4(a, bb, acc);
    }
    for (int j = 0; j < 8; ++j) {
      int d = mt * 16 + j + 8 * hi;
      dH[(size_t)blk * (Dd * Nn) + d * 64 + nt * 16 + nl] = acc[j];
    }
  }
}

// K5b: SEQUENTIAL state carry (130 short steps): per chunk
//   Yb += diag(Pvec) * Cc @ h^T   (WMMA)
//   h   = Ptot * h + dH           (elementwise, dH precomputed by K5a)
__global__ __launch_bounds__(256) void k_state_seq(const float* __restrict__ Cb,
                                                   const float* __restrict__ dH,
                                                   const float* __restrict__ Pvec,
                                                   const float* __restrict__ Ptot,
                                                   const float* __restrict__ hprev,
                                                   float* __restrict__ Yb,
                                                   float* __restrict__ hout) {
  __shared__ float hS[128 * 65];     // h[d][n]
  __shared__ float Cc[64 * 65];
  __shared__ float Pv[64];
  int b = blockIdx.x;
  int tid = threadIdx.x;
  int wave = tid >> 5, lane = tid & 31, ml = lane & 15, hi = lane >> 4, nl = ml;

  for (int i = tid; i < Dd * Nn; i += 256) {
    int d = i >> 6, n = i & 63;
    hS[d * 65 + n] = hprev[(size_t)b * Dd * Nn + i];
  }
  __syncthreads();

  for (int ch = 0; ch < NCH; ++ch) {
    int l0 = ch * CH;
    int blk = b * NCH + ch;
    for (int i = tid; i < CH * 16; i += 256) {
      int t = i >> 4, dg = (i & 15) * 4;
      async_ld_b128(lds_off(&Cc[t * 65 + dg]), Cb + ((size_t)b * Lf + l0 + t) * 64 + dg);
    }
    if (tid < 64) Pv[tid] = Pvec[(size_t)blk * 64 + tid];
    wait_async0();
    __syncthreads();

    // Yb += diag(Pv) Cc @ h^T : 4 x 8 tiles, K = 64 over n
    for (int q = wave * 4; q < wave * 4 + 4; ++q) {
      int mt = q >> 3, nt = q & 7;
      v8f acc = {0.f,0.f,0.f,0.f,0.f,0.f,0.f,0.f};
      float pscale = Pv[mt * 16 + ml];
      for (int k = 0; k < 64; k += 4) {
        int ks = k + 2 * hi;
        v2f a;  a.x  = pscale * Cc[(mt * 16 + ml) * 65 + ks];
                a.y  = pscale * Cc[(mt * 16 + ml) * 65 + ks + 1];
        v2f bb; bb.x = hS[(nt * 16 + nl) * 65 + ks];  bb.y = hS[(nt * 16 + nl) * 65 + ks + 1];
        acc = wmma4(a, bb, acc);
      }
      int dcol = nt * 16 + nl;
      for (int j = 0; j < 8; ++j) {
        int t = mt * 16 + j + 8 * hi;
        size_t yo = ((size_t)b * Dd + dcol) * Lf + l0 + t;
        Yb[yo] += acc[j];
      }
    }
    __syncthreads();                       // all reads of old h complete

    float pt = Ptot[blk];
    const float* dhc = dH + (size_t)blk * (Dd * Nn);
    for (int i = tid; i < Dd * Nn; i += 256) {
      int d = i >> 6, n = i & 63;
      hS[d * 65 + n] = pt * hS[d * 65 + n] + dhc[i];
    }
    __syncthreads();
  }

  for (int i = tid; i < Dd * Nn; i += 256) {
    int d = i >> 6, n = i & 63;
    hout[(size_t)b * Dd * Nn + i] = hS[d * 65 + n];
  }
}

// K6: output projection W_out @ y * sqrt_n; split into re/im frequency planes
__global__ __launch_bounds__(128) void k_out_gemm(const float* __restrict__ y,
                                                  const float* __restrict__ Wout,
                                                  float* __restrict__ ftre,
                                                  float* __restrict__ ftim) {
  __shared__ float Ys[128 * 16];
  int tt = blockIdx.x % (Lf / 16);
  int b  = blockIdx.x / (Lf / 16);
  int l0 = tt * 16;
  int tid = threadIdx.x;
  __builtin_prefetch(y + ((size_t)b * Dd + tid) * Lf + l0 + 16, 0, 0);
  for (int j = 0; j < 16; ++j) Ys[tid * 16 + j] = y[((size_t)b * Dd + tid) * Lf + l0 + j];
  __syncthreads();
  int wave = tid >> 5, lane = tid & 31, ml = lane & 15, hi = lane >> 4, nl = ml;
  for (int mt = wave; mt < 8; mt += 4) {
    int rowA = mt * 16 + ml;
    const float* wrow = Wout + (size_t)rowA * 128;
    v8f acc = {0.f,0.f,0.f,0.f,0.f,0.f,0.f,0.f};
    for (int k = 0; k < 128; k += 4) {
      int ks = k + 2 * hi;
      v2f a;  a.x  = wrow[ks];           a.y  = wrow[ks + 1];
      v2f bb; bb.x = Ys[ks * 16 + nl];   bb.y = Ys[(ks + 1) * 16 + nl];
      acc = wmma4(a, bb, acc);
    }
    int l = l0 + nl;
    for (int j = 0; j < 8; ++j) {
      int rowD = mt * 16 + j + 8 * hi;
      float v = acc[j] * 128.f;                       // * sqrt_n
      if (rowD < 64) ftre[((size_t)b * Cin + rowD) * Lf + l] = v;
      else           ftim[((size_t)b * Cin + rowD - 64) * Lf + l] = v;
    }
  }
}

// K7: inverse complex FFT along H per (b,c,w) column
__global__ __launch_bounds__(32) void k_ifft_cols(const float* __restrict__ ftre,
                                                  const float* __restrict__ ftim,
                                                  float* __restrict__ f2re,
                                                  float* __restrict__ f2im) {
  __shared__ float sre[128 * 32], sim[128 * 32];
  int q = blockIdx.x * 32 + threadIdx.x;              // (b,c,w)
  int w = q % Wf; int bc = q / Wf;
  float* re = sre + threadIdx.x; float* im = sim + threadIdx.x;
  const size_t base = (size_t)bc * Lf + w;
  for (int h = 0; h < 128; ++h) { re[h*32] = ftre[base + (size_t)h*Wf]; im[h*32] = ftim[base + (size_t)h*Wf]; }
  fft128_strided(re, im, 32, +1.f);
  for (int h = 0; h < 128; ++h) { f2re[base + (size_t)h*Wf] = re[h*32]; f2im[base + (size_t)h*Wf] = im[h*32]; }
}

// K8: Hermitian-extend 65 bins -> 128, inverse FFT along W, take real part, ortho scale
__global__ __launch_bounds__(32) void k_irfft_rows(const float* __restrict__ f2re,
                                                   const float* __restrict__ f2im,
                                                   float* __restrict__ out) {
  __shared__ float sre[128 * 32], sim[128 * 32];
  int r = blockIdx.x * 32 + threadIdx.x;              // (b,c,h)
  float* re = sre + threadIdx.x; float* im = sim + threadIdx.x;
  const size_t base = (size_t)r * Wf;
  for (int k = 0; k <= 64; ++k) { re[k*32] = f2re[base + k]; im[k*32] = f2im[base + k]; }
  for (int k = 65; k < 128; ++k) { re[k*32] = re[(128 - k)*32]; im[k*32] = -im[(128 - k)*32]; }
  fft128_strided(re, im, 32, +1.f);
  float* orow = out + (size_t)r * Ww;
  const float sc = 1.f / 128.f;                       // irfft2 ortho total scale
  for (int n = 0; n < 128; ++n) orow[n] = re[n*32] * sc;
}

extern "C" void kernel_launch(void* const* d_in, const int* in_sizes, int n_in,
                              void* d_out, int out_size, void* d_ws, size_t ws_size,
                              hipStream_t stream) {
  (void)in_sizes; (void)n_in; (void)out_size; (void)ws_size;
  const float* x      = (const float*)d_in[0];
  const float* hprev  = (const float*)d_in[1];
  const float* nw     = (const float*)d_in[2];
  const float* nb     = (const float*)d_in[3];
  const float* Wx     = (const float*)d_in[4];
  const float* Wb     = (const float*)d_in[5];
  const float* Wc     = (const float*)d_in[6];
  const float* wdt    = (const float*)d_in[7];
  const float* dtbias = (const float*)d_in[8];
  const float* alog   = (const float*)d_in[9];
  const float* Dskip  = (const float*)d_in[10];
  const float* Wout   = (const float*)d_in[11];

  float* out = (float*)d_out;
  float* ws  = (float*)d_ws;

  // Workspace layout (floats); total ~25.7M floats (~103 MB)
  size_t o = 0;
  float* Wre = ws + o; o += (size_t)Bn * Cin * Hh * Wf;   // row-FFT re  (reused as F2re)
  float* Wim = ws + o; o += (size_t)Bn * Cin * Hh * Wf;   // row-FFT im  (reused as F2im)
  float* U   = ws + o; o += (size_t)Bn * Dd * Lf;         // u / u-hat   (reused as FT planes)
  float* Xb  = ws + o; o += (size_t)Bn * Lf * 128;        // xs token-major
  float* Bb  = ws + o; o += (size_t)Bn * Lf * 64;
  float* Cb  = ws + o; o += (size_t)Bn * Lf * 64;
  float* dtb = ws + o; o += (size_t)Bn * Lf;
  float* Yb  = ws + o; o += (size_t)Bn * Dd * Lf;         // y = ys + D_skip*xs, [b][d][l]
  float* dH  = ws + o; o += (size_t)Bn * NCH * Dd * Nn;   // per-chunk state increments
  float* Pvec= ws + o; o += (size_t)Bn * NCH * CH;        // per-token chunk decay
  float* Ptot= ws + o; o += (size_t)Bn * NCH;             // total chunk decay

  float* FTre = U;                                        // U dead after proj GEMM
  float* FTim = U + (size_t)Bn * Cin * Lf;
  float* F2re = Wre;                                      // W planes dead after fwd FFT
  float* F2im = Wim;

  float* ysp  = out;
  float* hout = out + (size_t)Bn * Cin * Hh * Ww;

  k_rfft_rows   <<<(Bn * Cin * Hh) / 32, 32, 0, stream>>>(x, Wre, Wim);
  k_fft_cols_fwd<<<(Bn * Cin * Wf) / 32, 32, 0, stream>>>(Wre, Wim, U);
  k_ln_dt       <<<(Bn * Lf) / 32, 32, 0, stream>>>(U, nw, nb, wdt, dtbias, dtb);
  k_proj_gemm   <<<Bn * (Lf / 16), 128, 0, stream>>>(U, Wx, Wb, Wc, Xb, Bb, Cb);

  // parallel chunk phase: all 520 (b,chunk) blocks use the whole chip
  size_t smem = (size_t)(64 * 129 + 3 * 64 * 65 + 4 * 64) * sizeof(float);  // 83,968 B
  k_chunk_intra <<<Bn * NCH, 256, smem, stream>>>(Xb, Bb, Cb, dtb, Dskip, alog,
                                                  Yb, dH, Pvec, Ptot);
  // short sequential carry: 130 steps of one 64x128x64 WMMA GEMM + elementwise h update
  k_state_seq   <<<Bn, 256, 0, stream>>>(Cb, dH, Pvec, Ptot, hprev, Yb, hout);

  k_out_gemm    <<<Bn * (Lf / 16), 128, 0, stream>>>(Yb, Wout, FTre, FTim);
  k_ifft_cols   <<<(Bn * Cin * Wf) / 32, 32, 0, stream>>>(FTre, FTim, F2re, F2im);
  k_irfft_rows  <<<(Bn * Cin * Hh) / 32, 32, 0, stream>>>(F2re, F2im, ysp);
}